// EquivariantMessageLayer_68556267978768
// MI455X (gfx1250) — compile-verified
//
#include <hip/hip_runtime.h>

typedef __attribute__((ext_vector_type(2))) float v2f;
typedef __attribute__((ext_vector_type(8))) float v8f;
typedef __attribute__((ext_vector_type(4))) float f4;
typedef __attribute__((ext_vector_type(4))) unsigned int u32x4;
typedef __attribute__((ext_vector_type(8))) int i32x8;
typedef __attribute__((ext_vector_type(4))) int i32x4;

#define H    128
#define H3   384
#define ROWS 64          // node rows per workgroup (4 x 16-row WMMA subtiles)
#define LW   132         // LDS row stride in floats (128 + 4 pad via TDM pad)

// ---------------------------------------------------------------------------
// Kernel 1: s_expanded[N,384] = tanh(s @ W1 + b1) @ W2 + b2   (fp32 WMMA)
// 64 rows per workgroup. s tile staged to LDS by the Tensor Data Mover
// (tensor_load_to_lds) with padding -> stride-132 rows (bank-conflict free).
// Each B fragment load feeds 4 WMMAs (one per 16-row subtile).
// ---------------------------------------------------------------------------
__global__ __launch_bounds__(256) void embed_kernel(
    const float* __restrict__ s, const float* __restrict__ W1,
    const float* __restrict__ b1, const float* __restrict__ W2,
    const float* __restrict__ b2, float* __restrict__ s_exp, int N)
{
    __shared__ float s_lds[ROWS * LW];
    __shared__ float t_lds[ROWS * LW];

    const int tid  = threadIdx.x;
    const int lane = tid & 31;
    const int wv   = tid >> 5;          // wave 0..7
    const int row0 = blockIdx.x * ROWS;

    // ---- TDM: async-load the 64x128 f32 s tile into LDS (stride 132) ------
    if (wv == 0) {
        const uint64_t ga = (uint64_t)(uintptr_t)(s + (size_t)row0 * H);
        const uint32_t lb = (uint32_t)(uintptr_t)(void*)s_lds;
        const uint32_t rows_rem = (uint32_t)(N - row0);   // OOB rows read 0

        u32x4 g0;
        g0[0] = 1u;                                       // count=1 (valid D#)
        g0[1] = lb;                                       // lds_addr
        g0[2] = (uint32_t)ga;                             // global_addr[31:0]
        g0[3] = (uint32_t)((ga >> 32) & 0x1FFFFFFu)       // global_addr[56:32]
                | (2u << 30);                             // type=2 (image)

        i32x8 g1;
        g1[0] = (int)((2u << 16)        // data_size = 4 bytes
                    | (1u << 20)        // pad_enable
                    | (6u << 22)        // pad_interval: every 128 DWORDs
                    | (3u << 25));      // pad_amount: 4 DWORDs -> stride 132
        g1[1] = (int)(128u << 16);                        // tensor_dim0 = 128
        g1[2] = (int)(rows_rem << 16);                    // tensor_dim1 lo16
        g1[3] = (int)((rows_rem >> 16) | (128u << 16));   // dim1 hi | tile_dim0=128
        g1[4] = ROWS;                                     // tile_dim1 = 64
        g1[5] = 128;                                      // tensor_dim0_stride
        g1[6] = 0;
        g1[7] = 0;

        const i32x4 z4 = {};
        const i32x8 z8 = {};
        __builtin_amdgcn_tensor_load_to_lds(g0, g1, z4, z4, z8, 0);
        __builtin_amdgcn_s_wait_tensorcnt((short)0);
    }
    __syncthreads();

    const int half = lane >> 4;         // 0: K0/K1, 1: K2/K3
    const int ln   = lane & 15;         // A row / B,C,D column within tile

    // ---- stage 1: t = tanh(s @ W1 + b1); wave wv -> cols [16*wv, 16*wv+16)
    {
        const int col = wv * 16 + ln;
        v8f acc0 = {}, acc1 = {}, acc2 = {}, acc3 = {};
        #pragma unroll 4
        for (int k = 0; k < H; k += 4) {
            const int kb = k + 2 * half;
            v2f b;
            b.x = W1[(size_t)kb * H + col];
            b.y = W1[(size_t)(kb + 1) * H + col];
            v2f a0, a1, a2, a3;
            a0.x = s_lds[(ln +  0) * LW + kb]; a0.y = s_lds[(ln +  0) * LW + kb + 1];
            a1.x = s_lds[(ln + 16) * LW + kb]; a1.y = s_lds[(ln + 16) * LW + kb + 1];
            a2.x = s_lds[(ln + 32) * LW + kb]; a2.y = s_lds[(ln + 32) * LW + kb + 1];
            a3.x = s_lds[(ln + 48) * LW + kb]; a3.y = s_lds[(ln + 48) * LW + kb + 1];
            acc0 = __builtin_amdgcn_wmma_f32_16x16x4_f32(false, a0, false, b, (short)0, acc0, false, false);
            acc1 = __builtin_amdgcn_wmma_f32_16x16x4_f32(false, a1, false, b, (short)0, acc1, false, false);
            acc2 = __builtin_amdgcn_wmma_f32_16x16x4_f32(false, a2, false, b, (short)0, acc2, false, false);
            acc3 = __builtin_amdgcn_wmma_f32_16x16x4_f32(false, a3, false, b, (short)0, acc3, false, false);
        }
        const float bias = b1[col];
        #pragma unroll
        for (int r = 0; r < 8; ++r) {
            const int m = r + 8 * half;
            t_lds[(m +  0) * LW + col] = tanhf(acc0[r] + bias);
            t_lds[(m + 16) * LW + col] = tanhf(acc1[r] + bias);
            t_lds[(m + 32) * LW + col] = tanhf(acc2[r] + bias);
            t_lds[(m + 48) * LW + col] = tanhf(acc3[r] + bias);
        }
    }
    __syncthreads();

    // ---- stage 2: s_exp = t @ W2 + b2; wave wv -> col tiles 3*wv..3*wv+2
    for (int ot = 0; ot < 3; ++ot) {
        const int col = (wv * 3 + ot) * 16 + ln;
        v8f acc0 = {}, acc1 = {}, acc2 = {}, acc3 = {};
        #pragma unroll 4
        for (int k = 0; k < H; k += 4) {
            const int kb = k + 2 * half;
            v2f b;
            b.x = W2[(size_t)kb * H3 + col];
            b.y = W2[(size_t)(kb + 1) * H3 + col];
            v2f a0, a1, a2, a3;
            a0.x = t_lds[(ln +  0) * LW + kb]; a0.y = t_lds[(ln +  0) * LW + kb + 1];
            a1.x = t_lds[(ln + 16) * LW + kb]; a1.y = t_lds[(ln + 16) * LW + kb + 1];
            a2.x = t_lds[(ln + 32) * LW + kb]; a2.y = t_lds[(ln + 32) * LW + kb + 1];
            a3.x = t_lds[(ln + 48) * LW + kb]; a3.y = t_lds[(ln + 48) * LW + kb + 1];
            acc0 = __builtin_amdgcn_wmma_f32_16x16x4_f32(false, a0, false, b, (short)0, acc0, false, false);
            acc1 = __builtin_amdgcn_wmma_f32_16x16x4_f32(false, a1, false, b, (short)0, acc1, false, false);
            acc2 = __builtin_amdgcn_wmma_f32_16x16x4_f32(false, a2, false, b, (short)0, acc2, false, false);
            acc3 = __builtin_amdgcn_wmma_f32_16x16x4_f32(false, a3, false, b, (short)0, acc3, false, false);
        }
        const float bias = b2[col];
        #pragma unroll
        for (int r = 0; r < 8; ++r) {
            const int m = r + 8 * half;
            const int rr0 = row0 + m;
            if (rr0 < N)      s_exp[(size_t)rr0 * H3 + col]        = acc0[r] + bias;
            if (rr0+16 < N)   s_exp[(size_t)(rr0+16) * H3 + col]   = acc1[r] + bias;
            if (rr0+32 < N)   s_exp[(size_t)(rr0+32) * H3 + col]   = acc2[r] + bias;
            if (rr0+48 < N)   s_exp[(size_t)(rr0+48) * H3 + col]   = acc3[r] + bias;
        }
    }
}

// ---------------------------------------------------------------------------
// Kernel 0: zero accumulators
// ---------------------------------------------------------------------------
__global__ void zero_kernel(float* __restrict__ p, size_t n)
{
    size_t i = (size_t)blockIdx.x * blockDim.x + threadIdx.x;
    const size_t stride = (size_t)gridDim.x * blockDim.x;
    for (; i < n; i += stride) p[i] = 0.0f;
}

// ---------------------------------------------------------------------------
// Kernel 2: edge gather / message / scatter-add. One wave per edge.
// b128 gathers (all tables L2-resident), native f32 atomics for scatter.
// ---------------------------------------------------------------------------
__global__ __launch_bounds__(256) void edge_kernel(
    const float* __restrict__ v, const float* __restrict__ s_exp,
    const int* __restrict__ ei, const float* __restrict__ d_ij,
    const float* __restrict__ dir_ij, const float* __restrict__ Wd,
    const float* __restrict__ bd,
    float* __restrict__ dv_acc, float* __restrict__ ds_acc,
    float* __restrict__ cnt, int E)
{
    const int lane = threadIdx.x & 31;
    const int e = blockIdx.x * 8 + (threadIdx.x >> 5);
    if (e >= E) return;

    const int src = ei[e];
    const int dst = ei[E + e];
    const float d = d_ij[e];
    const float dir0 = dir_ij[3 * e + 0];
    const float dir1 = dir_ij[3 * e + 1];
    const float dir2 = dir_ij[3 * e + 2];

    const f4* se4 = (const f4*)(s_exp + (size_t)src * H3);
    const f4* vs4 = (const f4*)(v     + (size_t)src * H3);
    const f4* Wd4 = (const f4*)Wd;
    const f4* bd4 = (const f4*)bd;
    float* dvd = dv_acc + (size_t)dst * H3;
    float* dsd = ds_acc + (size_t)dst * H;

    const f4 gv = se4[lane]      * (d * Wd4[lane]      + bd4[lane]);       // dv_v
    const f4 gs = se4[32 + lane] * (d * Wd4[32 + lane] + bd4[32 + lane]);  // ds
    const f4 gr = se4[64 + lane] * (d * Wd4[64 + lane] + bd4[64 + lane]);  // dvr
    const f4 vv0 = vs4[lane], vv1 = vs4[32 + lane], vv2 = vs4[64 + lane];

    const int c = lane * 4;
    #pragma unroll
    for (int j = 0; j < 4; ++j) {
        __hip_atomic_fetch_add(&dsd[c + j], gs[j],
                               __ATOMIC_RELAXED, __HIP_MEMORY_SCOPE_AGENT);
        __hip_atomic_fetch_add(&dvd[c + j],          fmaf(gr[j], dir0, gv[j] * vv0[j]),
                               __ATOMIC_RELAXED, __HIP_MEMORY_SCOPE_AGENT);
        __hip_atomic_fetch_add(&dvd[H + c + j],      fmaf(gr[j], dir1, gv[j] * vv1[j]),
                               __ATOMIC_RELAXED, __HIP_MEMORY_SCOPE_AGENT);
        __hip_atomic_fetch_add(&dvd[2 * H + c + j],  fmaf(gr[j], dir2, gv[j] * vv2[j]),
                               __ATOMIC_RELAXED, __HIP_MEMORY_SCOPE_AGENT);
    }
    if (lane == 0)
        __hip_atomic_fetch_add(&cnt[dst], 1.0f,
                               __ATOMIC_RELAXED, __HIP_MEMORY_SCOPE_AGENT);
}

// ---------------------------------------------------------------------------
// Kernel 3: finalize.  out = [ v + dv_acc/denom  |  s + ds_acc/denom ]
// ---------------------------------------------------------------------------
__global__ __launch_bounds__(256) void finalize_kernel(
    const float* __restrict__ v, const float* __restrict__ s,
    const float* __restrict__ dv_acc, const float* __restrict__ ds_acc,
    const float* __restrict__ cnt, float* __restrict__ out, int N)
{
    const size_t idx = (size_t)blockIdx.x * blockDim.x + threadIdx.x;
    const size_t total_v = (size_t)N * H3;
    const size_t total   = total_v + (size_t)N * H;
    if (idx >= total) return;

    if (idx < total_v) {
        const int n = (int)(idx / H3);
        const float den = fmaxf(cnt[n], 1.0f);
        out[idx] = v[idx] + dv_acc[idx] / den;
    } else {
        const size_t j = idx - total_v;
        const int n = (int)(j / H);
        const float den = fmaxf(cnt[n], 1.0f);
        out[idx] = s[j] + ds_acc[j] / den;
    }
}

// ---------------------------------------------------------------------------
extern "C" void kernel_launch(void* const* d_in, const int* in_sizes, int n_in,
                              void* d_out, int out_size, void* d_ws, size_t ws_size,
                              hipStream_t stream)
{
    const float* v      = (const float*)d_in[0];
    const float* s      = (const float*)d_in[1];
    const int*   ei     = (const int*)  d_in[2];
    const float* d_ij   = (const float*)d_in[3];
    const float* dir_ij = (const float*)d_in[4];
    const float* W1     = (const float*)d_in[5];
    const float* b1     = (const float*)d_in[6];
    const float* W2     = (const float*)d_in[7];
    const float* b2     = (const float*)d_in[8];
    const float* Wd     = (const float*)d_in[9];
    const float* bd     = (const float*)d_in[10];

    const int N = in_sizes[1] / H;     // 50000
    const int E = in_sizes[3];         // 500000

    // workspace (floats): s_exp[N*384] | dv_acc[N*384] | ds_acc[N*128] | cnt[N]
    float* ws     = (float*)d_ws;
    float* s_exp  = ws;
    float* dv_acc = s_exp  + (size_t)N * H3;
    float* ds_acc = dv_acc + (size_t)N * H3;
    float* cnt    = ds_acc + (size_t)N * H;

    // 0) zero accumulators
    const size_t nz = (size_t)N * (H3 + H + 1);
    zero_kernel<<<2048, 256, 0, stream>>>(dv_acc, nz);

    // 1) embed GEMM (WMMA fp32 + TDM tile staging)
    const int nblk = (N + ROWS - 1) / ROWS;
    embed_kernel<<<nblk, 256, 0, stream>>>(s, W1, b1, W2, b2, s_exp, N);

    // 2) edge gather/scatter (one wave per edge)
    edge_kernel<<<(E + 7) / 8, 256, 0, stream>>>(v, s_exp, ei, d_ij, dir_ij,
                                                 Wd, bd, dv_acc, ds_acc, cnt, E);

    // 3) finalize into d_out
    const size_t total = (size_t)N * (H3 + H);
    finalize_kernel<<<(int)((total + 255) / 256), 256, 0, stream>>>(
        v, s, dv_acc, ds_acc, cnt, (float*)d_out, N);
}